// BeamSearch_19877108646657
// MI455X (gfx1250) — compile-verified
//
#include <hip/hip_runtime.h>
#include <hip/hip_bf16.h>
#include <stdint.h>

// Problem constants (static in the reference)
#define N_   64
#define KP_  16
#define V_   32000
#define TM1_ 63
#define K_   16

// ---------------------------------------------------------------------------
// CDNA5 async global->LDS copy path (ASYNCcnt-tracked), with safe fallbacks.
// Probe result: builtin exists and takes pointers to 16-byte int vectors.
// ---------------------------------------------------------------------------
#if defined(__AMDGCN__) && __has_builtin(__builtin_amdgcn_global_load_async_to_lds_b128)
#define HAVE_ASYNC_LDS 1
#else
#define HAVE_ASYNC_LDS 0
#endif

#if HAVE_ASYNC_LDS
typedef int v4i_vs __attribute__((vector_size(16)));
typedef __attribute__((address_space(1))) v4i_vs* gptr_v4i;
typedef __attribute__((address_space(3))) v4i_vs* lptr_v4i;
#endif

__device__ __forceinline__ void async_copy_b128(const float* gsrc, float* ldst) {
#if HAVE_ASYNC_LDS
  __builtin_amdgcn_global_load_async_to_lds_b128(
      (gptr_v4i)gsrc, (lptr_v4i)ldst, /*offset=*/0, /*cpol=*/0);
#else
  *(float4*)ldst = *(const float4*)gsrc;
#endif
}

template <int Cnt>
__device__ __forceinline__ void wait_async() {
#if HAVE_ASYNC_LDS
#if __has_builtin(__builtin_amdgcn_s_wait_asynccnt)
  __builtin_amdgcn_s_wait_asynccnt(Cnt);
#else
  asm volatile("s_wait_asynccnt %0" :: "n"(Cnt) : "memory");
#endif
  asm volatile("" ::: "memory");
#endif
}

// ---------------------------------------------------------------------------
// Merge two sorted-descending 16-lists (in LDS), keep top-16 into (av, ai).
// ---------------------------------------------------------------------------
__device__ __forceinline__ void merge_top16(float* av, int* ai,
                                            const float* bv, const int* bi) {
  float tv[K_]; int ti[K_];
  int ia = 0, ib = 0;
#pragma unroll
  for (int r = 0; r < K_; ++r) {
    float x = av[ia], y = bv[ib];
    if (x >= y) { tv[r] = x; ti[r] = ai[ia]; ++ia; }
    else        { tv[r] = y; ti[r] = bi[ib]; ++ib; }
  }
#pragma unroll
  for (int r = 0; r < K_; ++r) { av[r] = tv[r]; ai[r] = ti[r]; }
}

// ---------------------------------------------------------------------------
// Kernel 1: per (n,kp) row, stream 32000 floats through LDS with async copies,
// keep per-thread register top-16, reduce to row top-16 in workspace.
// ---------------------------------------------------------------------------
__global__ __launch_bounds__(256) void bs_rowtopk(
    const float* __restrict__ lp_t, const float* __restrict__ lp_prev,
    float* __restrict__ wsv, int* __restrict__ wsi) {
  __shared__ __align__(16) unsigned char smem[32768];
  float* stage = (float*)smem;          // 2 x 1024 floats (8KB), streaming phase
  float* rv    = (float*)smem;          // 256*16 floats (16KB), reduce phase
  int*   ri    = (int*)(smem + 16384);  // 256*16 ints   (16KB), reduce phase

  const int tid = threadIdx.x;
  const int row = blockIdx.x;           // row = n*KP + kp
  const int kp  = row & (KP_ - 1);
  const float bias = lp_prev[row];
  const float* __restrict__ rowp = lp_t + (size_t)row * V_;

  constexpr int CHUNK  = 1024;                       // floats per chunk
  constexpr int NCHUNK = (V_ + CHUNK - 1) / CHUNK;   // 32 (last partial: 256)

  float vals[K_]; int idxs[K_];
#pragma unroll
  for (int j = 0; j < K_; ++j) { vals[j] = -3.402823466e38f; idxs[j] = 0; }

  // Prologue: chunk 0 in flight.
  if (tid * 4 < V_) async_copy_b128(rowp + tid * 4, stage + tid * 4);

  for (int c = 0; c < NCHUNK; ++c) {
    if (c + 1 < NCHUNK) {
      const int g = (c + 1) * CHUNK + tid * 4;
      if (g < V_)
        async_copy_b128(rowp + g, stage + ((c + 1) & 1) * CHUNK + tid * 4);
      wait_async<1>();   // oldest (chunk c) landed; next chunk still in flight
    } else {
      wait_async<0>();
    }
    const int base = c * CHUNK + tid * 4;
    float4 q = *(const float4*)(stage + (c & 1) * CHUNK + tid * 4);
    const float qv[4] = {q.x, q.y, q.z, q.w};
#pragma unroll
    for (int j = 0; j < 4; ++j) {
      if (base + j < V_) {
        const float v = qv[j] + bias;
        if (v > vals[K_ - 1]) {
          vals[K_ - 1] = v;
          idxs[K_ - 1] = kp * V_ + base + j;
#pragma unroll
          for (int t = K_ - 1; t > 0; --t) {
            if (vals[t] > vals[t - 1]) {
              float tv = vals[t]; vals[t] = vals[t - 1]; vals[t - 1] = tv;
              int   ti = idxs[t]; idxs[t] = idxs[t - 1]; idxs[t - 1] = ti;
            }
          }
        }
      }
    }
  }

  __syncthreads();  // staging region is reused below as reduction arrays
#pragma unroll
  for (int j = 0; j < K_; ++j) { rv[tid * K_ + j] = vals[j]; ri[tid * K_ + j] = idxs[j]; }
  __syncthreads();

  // Pairwise merge tree: 256 sorted 16-lists -> block top-16 at slot 0.
  for (int off = 1; off < 256; off <<= 1) {
    if ((tid & (2 * off - 1)) == 0)
      merge_top16(&rv[tid * K_], &ri[tid * K_],
                  &rv[(tid + off) * K_], &ri[(tid + off) * K_]);
    __syncthreads();
  }

  if (tid < K_) {
    wsv[row * K_ + tid] = rv[tid];
    wsi[row * K_ + tid] = ri[tid];
  }
}

// ---------------------------------------------------------------------------
// Kernel 2: per batch n, merge the 16 row-candidate lists -> global top-16,
// emit (lens, log_probs, src), then gather prefixes + scatter new tokens.
// ---------------------------------------------------------------------------
__global__ __launch_bounds__(256) void bs_finalize(
    const float* __restrict__ wsv, const int* __restrict__ wsi,
    const int* __restrict__ y_prev, const int* __restrict__ y_prev_lens,
    int* __restrict__ out_y, int* __restrict__ out_lens,
    float* __restrict__ out_lp, int* __restrict__ out_src) {
  __shared__ float fv[KP_ * K_];
  __shared__ int   fi[KP_ * K_];
  __shared__ int   s_src[K_], s_len[K_], s_yt[K_];

  const int tid = threadIdx.x;
  const int n   = blockIdx.x;

  if (tid < KP_ * K_) {
    fv[tid] = wsv[n * KP_ * K_ + tid];
    fi[tid] = wsi[n * KP_ * K_ + tid];
  }
  __syncthreads();

  for (int off = 1; off < KP_; off <<= 1) {
    if (tid < KP_ && (tid & (2 * off - 1)) == 0)
      merge_top16(&fv[tid * K_], &fi[tid * K_],
                  &fv[(tid + off) * K_], &fi[(tid + off) * K_]);
    __syncthreads();
  }

  if (tid < K_) {
    const float v   = fv[tid];
    const int  flat = fi[tid];
    const int  src  = flat / V_;
    const int  yt   = flat - src * V_;
    const int  len  = y_prev_lens[n * KP_ + src];
    out_lens[n * K_ + tid] = len + 1;
    out_lp  [n * K_ + tid] = v;
    out_src [n * K_ + tid] = src;
    s_src[tid] = src; s_len[tid] = len; s_yt[tid] = yt;
  }
  __syncthreads();

  // y_next[t, n, r] = y_prev[t, n, src[r]]; y_next[len[r], n, r] = yt[r]
  for (int e = tid; e < TM1_ * K_; e += 256) {
    const int t = e >> 4, r = e & (K_ - 1);
    const int srcb = s_src[r];
    int v = y_prev[((size_t)t * N_ + n) * KP_ + srcb];
    if (t == s_len[r]) v = s_yt[r];
    out_y[((size_t)t * N_ + n) * K_ + r] = v;
  }
}

// ---------------------------------------------------------------------------
// Host launcher
// ---------------------------------------------------------------------------
extern "C" void kernel_launch(void* const* d_in, const int* in_sizes, int n_in,
                              void* d_out, int out_size, void* d_ws, size_t ws_size,
                              hipStream_t stream) {
  (void)in_sizes; (void)n_in; (void)out_size; (void)ws_size;

  const float* lp_t     = (const float*)d_in[0];  // (64,16,32000) f32
  /* d_in[1] = width (static 16, K_ == WIDTH) */
  const float* lp_prev  = (const float*)d_in[2];  // (64,16) f32
  const int*   y_prev   = (const int*)d_in[3];    // (63,64,16) i32
  const int*   yp_lens  = (const int*)d_in[4];    // (64,16) i32

  // Workspace: per-row top-16 (values then indices) = 1024*16 of each.
  float* wsv = (float*)d_ws;
  int*   wsi = (int*)((char*)d_ws + (size_t)N_ * KP_ * K_ * sizeof(float));

  // Output layout (flat, return order):
  //   y_next        : 63*64*16 = 64512 ints  @ [0)
  //   y_next_lens   : 1024 ints              @ [64512)
  //   log_probs_next: 1024 floats            @ [65536)
  //   next_src      : 1024 ints              @ [66560)
  int*   out_i = (int*)d_out;
  float* out_f = (float*)d_out;

  bs_rowtopk<<<N_ * KP_, 256, 0, stream>>>(lp_t, lp_prev, wsv, wsi);
  bs_finalize<<<N_, 256, 0, stream>>>(wsv, wsi, y_prev, yp_lens,
                                      out_i,            // y_next
                                      out_i + 64512,    // y_next_lens
                                      out_f + 65536,    // log_probs_next
                                      out_i + 66560);   // next_src
}